// Tacotron2Attention_78649441124506
// MI455X (gfx1250) — compile-verified
//
#include <hip/hip_runtime.h>
#include <math.h>

// Problem constants (match reference)
#define B_     128
#define S_     1024
#define FILT_  32
#define K_     31
#define PAD_   15
#define ATT_   128
#define RNN2_  2048
#define ENC_   512

typedef __attribute__((ext_vector_type(2))) float v2f;
typedef __attribute__((ext_vector_type(4))) float v4f;
typedef __attribute__((ext_vector_type(8))) float v8f;

// ---------------------------------------------------------------------------
// Kernel 1: q[b,d] = sum_i h[b,i] * W_rnn[i,d],  h = concat(rnn_hs[0], rnn_hs[1])
// grid = B, block = 128 (one thread per d). h cached in LDS (broadcast reads),
// W_rnn reads coalesced across d and L2-resident (1 MB).
// ---------------------------------------------------------------------------
__global__ void query_kernel(const float* __restrict__ rnn_hs,
                             const float* __restrict__ W_rnn,
                             float* __restrict__ qbuf) {
    __shared__ float lds_h[RNN2_];
    const int b = blockIdx.x;
    const int tid = threadIdx.x;
    for (int i = tid; i < 1024; i += 128) {
        lds_h[i]        = rnn_hs[(size_t)b * 1024 + i];                     // rnn_hs[0]
        lds_h[1024 + i] = rnn_hs[(size_t)B_ * 1024 + (size_t)b * 1024 + i]; // rnn_hs[1]
    }
    __syncthreads();
    float acc = 0.f;
    const int d = tid;
#pragma unroll 4
    for (int i = 0; i < RNN2_; ++i) {
        acc = fmaf(lds_h[i], W_rnn[(size_t)i * ATT_ + d], acc);
    }
    qbuf[(size_t)b * ATT_ + d] = acc;
}

// ---------------------------------------------------------------------------
// Kernel 2 (WMMA): per 16-row S tile of one batch element:
//   1) conv1d taps -> loc tile (16 x 32) in LDS
//   2) loc @ W_loc_lin via V_WMMA_F32_16X16X4_F32 (one wave per 16-col N tile)
//   3) fused + q[b] + processed_encoder_output (NT stream), tanh, * W_align,
//      butterfly-reduce across the 16 N columns of each wave tile,
//      atomicAdd into lds_e[16], apply mask, emit e[b,s].
// grid = B*64 blocks, block = 256 threads (8 waves).
// ---------------------------------------------------------------------------
__global__ void energy_kernel(const float* __restrict__ loc_att,      // (B,S,1)
                              const unsigned char* __restrict__ mask, // (B,S) bool
                              const float* __restrict__ W_loc,        // (32,1,31)
                              const float* __restrict__ W_loc_lin,    // (32,128)
                              const float* __restrict__ W_align,      // (128)
                              const float* __restrict__ pe,           // (B,S,128)
                              const float* __restrict__ qbuf,         // (B,128)
                              float* __restrict__ e_out) {            // (B,S)
    __shared__ float lds_att[48];            // padded conv window (46 used)
    __shared__ float lds_wloc[FILT_ * K_];   // 992
    __shared__ float lds_loc[16 * 33];       // 16 rows, pitch 33 (bank friendly)
    __shared__ float lds_e[16];

    const int b    = blockIdx.x >> 6;        // B*64 blocks
    const int tile = blockIdx.x & 63;
    const int s0   = tile * 16;
    const int tid  = threadIdx.x;

    // --- stage conv inputs + weights ---
    if (tid < 46) {
        int g = s0 - PAD_ + tid;
        lds_att[tid] = (g >= 0 && g < S_) ? loc_att[(size_t)b * S_ + g] : 0.f;
    }
    if (tid < 16) lds_e[tid] = 0.f;
    for (int i = tid; i < FILT_ * K_; i += 256) lds_wloc[i] = W_loc[i];
    __syncthreads();

    // --- conv: out[m][f] = sum_k W_loc[f,k] * att[s0+m+k-15] ---
    for (int idx = tid; idx < 16 * FILT_; idx += 256) {
        const int m = idx >> 5;
        const int f = idx & 31;
        float acc = 0.f;
#pragma unroll
        for (int k = 0; k < K_; ++k)
            acc = fmaf(lds_wloc[f * K_ + k], lds_att[m + k], acc);
        lds_loc[m * 33 + f] = acc;
    }
    __syncthreads();

    // --- WMMA loc(16x32) @ W_loc_lin(32x128): wave w handles N-tile n0=w*16 ---
    const int wave = tid >> 5;
    const int lane = tid & 31;
    const int half = lane >> 4;   // 0: lanes 0-15, 1: lanes 16-31
    const int nloc = lane & 15;
    const int n0   = wave * 16;

    v8f acc = {};
#pragma unroll
    for (int kstep = 0; kstep < 8; ++kstep) {
        const int k0 = kstep * 4;
        // A 16x4 f32 fragment: lanes hold rows M=lane&15; VGPR0/1 = K {0,1} (half 0) / {2,3} (half 1)
        v2f a;
        a.x = lds_loc[nloc * 33 + k0 + 2 * half + 0];
        a.y = lds_loc[nloc * 33 + k0 + 2 * half + 1];
        // B 4x16 f32 fragment: lanes hold cols N=n0+nloc; same K striping
        v2f bb;
        bb.x = W_loc_lin[(size_t)(k0 + 2 * half + 0) * ATT_ + n0 + nloc];
        bb.y = W_loc_lin[(size_t)(k0 + 2 * half + 1) * ATT_ + n0 + nloc];
        acc = __builtin_amdgcn_wmma_f32_16x16x4_f32(false, a, false, bb,
                                                    (short)0, acc, false, false);
    }

    // --- D layout: acc[r] = D[m = r + 8*half][n = n0 + nloc]
    //     fused add + tanh + * W_align, reduce over 16 N columns ---
    const float qv  = qbuf[(size_t)b * ATT_ + n0 + nloc];
    const float wal = W_align[n0 + nloc];
#pragma unroll
    for (int r = 0; r < 8; ++r) {
        const int m = r + 8 * half;
        const int s = s0 + m;
        // pe is 64 MB streamed exactly once -> non-temporal load (TH=NT)
        const float pev =
            __builtin_nontemporal_load(&pe[((size_t)b * S_ + s) * ATT_ + n0 + nloc]);
        float val = tanhf(acc[r] + qv + pev) * wal;
        // butterfly over the 16-lane group (masks < 16 keep halves separate)
        val += __shfl_xor(val, 1, 32);
        val += __shfl_xor(val, 2, 32);
        val += __shfl_xor(val, 4, 32);
        val += __shfl_xor(val, 8, 32);
        if (nloc == 0) atomicAdd(&lds_e[m], val);  // ds_add_f32
    }
    __syncthreads();

    if (tid < 16) {
        const int s = s0 + tid;
        float ev = lds_e[tid];
        if (!mask[(size_t)b * S_ + s]) ev = -INFINITY;
        e_out[(size_t)b * S_ + s] = ev;
    }
}

// ---------------------------------------------------------------------------
// Kernel 3: softmax over S per batch row; writes alignments into d_out.
// grid = B, block = 256, 4 elements/thread (coalesced, stride 256).
// ---------------------------------------------------------------------------
__global__ void softmax_kernel(const float* __restrict__ e_in,
                               float* __restrict__ align_out) {
    __shared__ float red[8];
    const int b = blockIdx.x;
    const int tid = threadIdx.x;
    const int wid = tid >> 5;
    const int lane = tid & 31;

    float v[4];
#pragma unroll
    for (int j = 0; j < 4; ++j)
        v[j] = e_in[(size_t)b * S_ + tid + 256 * j];

    // max reduce
    float m = fmaxf(fmaxf(v[0], v[1]), fmaxf(v[2], v[3]));
#pragma unroll
    for (int off = 16; off >= 1; off >>= 1)
        m = fmaxf(m, __shfl_xor(m, off, 32));
    if (lane == 0) red[wid] = m;
    __syncthreads();
    if (tid == 0) {
        float g = red[0];
#pragma unroll
        for (int i = 1; i < 8; ++i) g = fmaxf(g, red[i]);
        red[0] = g;
    }
    __syncthreads();
    const float gmax = red[0];
    __syncthreads();

    // exp + sum reduce
    float s = 0.f;
#pragma unroll
    for (int j = 0; j < 4; ++j) {
        v[j] = __expf(v[j] - gmax);
        s += v[j];
    }
#pragma unroll
    for (int off = 16; off >= 1; off >>= 1)
        s += __shfl_xor(s, off, 32);
    if (lane == 0) red[wid] = s;
    __syncthreads();
    if (tid == 0) {
        float g = 0.f;
#pragma unroll
        for (int i = 0; i < 8; ++i) g += red[i];
        red[0] = g;
    }
    __syncthreads();
    const float inv = 1.f / red[0];

#pragma unroll
    for (int j = 0; j < 4; ++j)
        align_out[(size_t)b * S_ + tid + 256 * j] = v[j] * inv;
}

// ---------------------------------------------------------------------------
// Kernel 4: partial context sums. grid = (B, 8); block = 128 threads, each
// owning 4 consecutive channels (float4 -> global_load_b128). encoder_output
// is 256 MB (> 192 MB L2) streamed exactly once -> non-temporal loads so the
// stream doesn't evict reused weights/alignments from L2.
// part[(b*8+j)*512 + c..c+3] = sum_{s in chunk j} enc[b,s,c..c+3] * align[b,s]
// ---------------------------------------------------------------------------
__global__ void context_partial(const float* __restrict__ enc,
                                const float* __restrict__ align,
                                float* __restrict__ part) {
    const int b = blockIdx.x;
    const int j = blockIdx.y;
    const int c4 = threadIdx.x;            // float4 index: channels c4*4 .. c4*4+3
    const int sbeg = j * 128;

    const v4f* encv = (const v4f*)enc;     // ENC_/4 = 128 float4 per (b,s) row

    v4f acc = {0.f, 0.f, 0.f, 0.f};
#pragma unroll 4
    for (int s = sbeg; s < sbeg + 128; ++s) {
        const float a = align[(size_t)b * S_ + s];
        const v4f ev =
            __builtin_nontemporal_load(&encv[((size_t)b * S_ + s) * (ENC_ / 4) + c4]);
        acc.x = fmaf(ev.x, a, acc.x);
        acc.y = fmaf(ev.y, a, acc.y);
        acc.z = fmaf(ev.z, a, acc.z);
        acc.w = fmaf(ev.w, a, acc.w);
    }
    v4f* partv = (v4f*)part;
    partv[((size_t)b * 8 + j) * (ENC_ / 4) + c4] = acc;
}

// Kernel 5: final reduction over the 8 chunks -> attention_score (overwrite).
__global__ void context_reduce(const float* __restrict__ part,
                               float* __restrict__ score) {
    const int b = blockIdx.x;
    const int c = threadIdx.x;
    float acc = 0.f;
#pragma unroll
    for (int j = 0; j < 8; ++j)
        acc += part[((size_t)b * 8 + j) * ENC_ + c];
    score[(size_t)b * ENC_ + c] = acc;
}

// ---------------------------------------------------------------------------
// Host launcher
// ---------------------------------------------------------------------------
extern "C" void kernel_launch(void* const* d_in, const int* in_sizes, int n_in,
                              void* d_out, int out_size, void* d_ws, size_t ws_size,
                              hipStream_t stream) {
    const float* rnn_hs   = (const float*)d_in[0];             // (2,B,1024)
    const float* enc      = (const float*)d_in[1];             // (B,S,512)
    const float* pe       = (const float*)d_in[2];             // (B,S,128)
    const float* loc_att  = (const float*)d_in[3];             // (B,S,1)
    const unsigned char* mask = (const unsigned char*)d_in[4]; // (B,S) bool
    const float* W_loc     = (const float*)d_in[5];            // (32,1,31)
    const float* W_loc_lin = (const float*)d_in[6];            // (32,128)
    const float* W_rnn     = (const float*)d_in[7];            // (2048,128)
    const float* W_align   = (const float*)d_in[8];            // (128,)

    // d_out: [ attention_score (B*ENC) | alignments (B*S) ]
    float* out_score = (float*)d_out;
    float* out_align = (float*)d_out + (size_t)B_ * ENC_;

    // workspace layout (floats): e (B*S) | q (B*ATT) | partials (B*8*ENC)
    float* ws_e    = (float*)d_ws;
    float* ws_q    = ws_e + (size_t)B_ * S_;
    float* ws_part = ws_q + (size_t)B_ * ATT_;

    query_kernel<<<B_, 128, 0, stream>>>(rnn_hs, W_rnn, ws_q);

    energy_kernel<<<B_ * 64, 256, 0, stream>>>(loc_att, mask, W_loc, W_loc_lin,
                                               W_align, pe, ws_q, ws_e);

    softmax_kernel<<<B_, 256, 0, stream>>>(ws_e, out_align);

    context_partial<<<dim3(B_, 8), 128, 0, stream>>>(enc, out_align, ws_part);

    context_reduce<<<B_, ENC_, 0, stream>>>(ws_part, out_score);
}